// GraphClassificationModelBinaryCrossEntropy_66288525247279
// MI455X (gfx1250) — compile-verified
//
#include <hip/hip_runtime.h>
#include <hip/hip_bf16.h>

typedef __attribute__((ext_vector_type(2))) float v2f;
typedef __attribute__((ext_vector_type(8))) float v8f;

#define N_NODES 100000
#define N_EDGES 3200000
#define IN_DIM 256
#define OUT_DIM 16

// ---------------- zero init (degrees + output accumulator) ----------------
__global__ void gcn_zero(float* __restrict__ deg, int ndeg,
                         float* __restrict__ out, int nout) {
    int i = blockIdx.x * blockDim.x + threadIdx.x;
    int total = ndeg + nout;
    for (; i < total; i += gridDim.x * blockDim.x) {
        if (i < ndeg) deg[i] = 0.0f;
        else          out[i - ndeg] = 0.0f;
    }
}

// ---------------- degree computation: 2 f32 atomics per edge ----------------
__global__ void gcn_degrees(const int* __restrict__ src, const int* __restrict__ dst,
                            float* __restrict__ outdeg, float* __restrict__ indeg,
                            int n_edges) {
    int e = blockIdx.x * blockDim.x + threadIdx.x;
    if (e >= n_edges) return;
    unsafeAtomicAdd(&outdeg[src[e]], 1.0f);
    unsafeAtomicAdd(&indeg[dst[e]], 1.0f);
}

// ---------------- WMMA GEMM: h = (X * outnorm) @ W ----------------
// One wave32 computes one 16x16 output tile: 16 node rows x 16 out dims.
// K = 256 consumed 4 at a time by v_wmma_f32_16x16x4_f32 (64 WMMAs/wave).
__global__ __launch_bounds__(256) void gcn_gemm_wmma(
    const float* __restrict__ X,       // [N, 256]
    const float* __restrict__ outdeg,  // [N]
    const float* __restrict__ Wm,      // [256, 16]
    float* __restrict__ H,             // [N, 16]
    int n_nodes) {
    __shared__ float Ws[IN_DIM * OUT_DIM];  // 16 KB
    for (int i = threadIdx.x; i < IN_DIM * OUT_DIM; i += blockDim.x)
        Ws[i] = Wm[i];
    __syncthreads();

    const int lane = threadIdx.x & 31;
    const int wave = threadIdx.x >> 5;
    const int tile = blockIdx.x * 8 + wave;   // 16-row tile per wave
    const int row0 = tile * 16;
    if (row0 >= n_nodes) return;              // wave-uniform: EXEC stays all-1s

    // A fragment addressing (16x4 f32): lanes 0-15 -> K={0,1}, lanes 16-31 -> K={2,3}
    const int m  = lane & 15;                 // row within tile; also N for B/C
    const int kb = (lane >> 4) * 2;           // K sub-offset: 0 or 2
    const int row = row0 + m;                 // 100000 % 16 == 0, always in range

    float nrm = __frsqrt_rn(fmaxf(outdeg[row], 1.0f));
    const float* xrow = X + (size_t)row * IN_DIM;

    v8f c = {};
#pragma unroll
    for (int k0 = 0; k0 < IN_DIM; k0 += 4) {
        v2f a, b;
        a.x = xrow[k0 + kb] * nrm;
        a.y = xrow[k0 + kb + 1] * nrm;
        b.x = Ws[(k0 + kb) * OUT_DIM + m];
        b.y = Ws[(k0 + kb + 1) * OUT_DIM + m];
        // (neg_a, A, neg_b, B, c_mod, C, reuse_a, reuse_b)
        c = __builtin_amdgcn_wmma_f32_16x16x4_f32(false, a, false, b,
                                                  (short)0, c, false, false);
    }

    // C/D layout: VGPR r holds M=r (lanes 0-15) and M=r+8 (lanes 16-31), N=lane&15
    const int half = lane >> 4;
#pragma unroll
    for (int r = 0; r < 8; ++r) {
        int orow = row0 + r + half * 8;
        H[(size_t)orow * OUT_DIM + m] = c[r];
    }
}

// ---------------- scatter: out[dst] += h[src], 16 f32 atomics/edge ----------------
__global__ void gcn_scatter(const int* __restrict__ src, const int* __restrict__ dst,
                            const float* __restrict__ H, float* __restrict__ out,
                            int n_edges) {
    int e = blockIdx.x * blockDim.x + threadIdx.x;
    if (e >= n_edges) return;
    const int s = src[e], d = dst[e];
    const float4* hp = (const float4*)(H + (size_t)s * OUT_DIM);
    float* op = out + (size_t)d * OUT_DIM;
#pragma unroll
    for (int q = 0; q < 4; ++q) {
        float4 v = hp[q];
        unsafeAtomicAdd(op + q * 4 + 0, v.x);
        unsafeAtomicAdd(op + q * 4 + 1, v.y);
        unsafeAtomicAdd(op + q * 4 + 2, v.z);
        unsafeAtomicAdd(op + q * 4 + 3, v.w);
    }
}

// ---------------- finalize: out = out * innorm + b ----------------
__global__ void gcn_finalize(float* __restrict__ out, const float* __restrict__ indeg,
                             const float* __restrict__ bias, int n_nodes) {
    int i = blockIdx.x * blockDim.x + threadIdx.x;
    if (i >= n_nodes) return;
    float nrm = __frsqrt_rn(fmaxf(indeg[i], 1.0f));
    float4* op = (float4*)(out + (size_t)i * OUT_DIM);
    const float4* bp = (const float4*)bias;
#pragma unroll
    for (int q = 0; q < 4; ++q) {
        float4 v = op[q];
        float4 bb = bp[q];
        v.x = v.x * nrm + bb.x;
        v.y = v.y * nrm + bb.y;
        v.z = v.z * nrm + bb.z;
        v.w = v.w * nrm + bb.w;
        op[q] = v;
    }
}

extern "C" void kernel_launch(void* const* d_in, const int* in_sizes, int n_in,
                              void* d_out, int out_size, void* d_ws, size_t ws_size,
                              hipStream_t stream) {
    const float* features = (const float*)d_in[0];  // [100000, 256]
    const int*   src      = (const int*)d_in[1];    // [3.2M]
    const int*   dst      = (const int*)d_in[2];    // [3.2M]
    const float* W        = (const float*)d_in[3];  // [256, 16]
    const float* b        = (const float*)d_in[4];  // [16]
    float* out = (float*)d_out;                     // [100000, 16]

    // workspace layout: outdeg[N] | indeg[N] | h[N*16]
    float* ws     = (float*)d_ws;
    float* outdeg = ws;
    float* indeg  = ws + N_NODES;
    float* h      = ws + 2 * N_NODES;

    const int n_nodes = in_sizes[0] / IN_DIM;   // 100000
    const int n_edges = in_sizes[1];            // 3200000

    // 1) zero degrees + output accumulator
    {
        int ndeg = 2 * n_nodes;
        int nout = n_nodes * OUT_DIM;
        int total = ndeg + nout;
        int blocks = (total + 255) / 256;
        gcn_zero<<<blocks, 256, 0, stream>>>(outdeg, ndeg, out, nout);
    }
    // 2) degrees
    gcn_degrees<<<(n_edges + 255) / 256, 256, 0, stream>>>(src, dst, outdeg, indeg, n_edges);
    // 3) projection GEMM with WMMA (one wave per 16-row tile, 8 waves/block)
    {
        int tiles = (n_nodes + 15) / 16;        // 6250
        int blocks = (tiles + 7) / 8;           // 782
        gcn_gemm_wmma<<<blocks, 256, 0, stream>>>(features, outdeg, W, h, n_nodes);
    }
    // 4) edge scatter-add
    gcn_scatter<<<(n_edges + 255) / 256, 256, 0, stream>>>(src, dst, h, out, n_edges);
    // 5) finalize
    gcn_finalize<<<(n_nodes + 255) / 256, 256, 0, stream>>>(out, indeg, b, n_nodes);
}